// A2_Projection_17824114278487
// MI455X (gfx1250) — compile-verified
//
#include <hip/hip_runtime.h>
#include <math.h>

typedef __attribute__((ext_vector_type(2))) float v2f;
typedef __attribute__((ext_vector_type(4))) float v4f;
typedef __attribute__((ext_vector_type(8))) float v8f;

#define B_ 8
#define C_ 256
#define N_ 9216
#define K_ 64
#define S_ 256
#define EPS_BN 1e-5f
#define EPS_NORM 1e-12f
#define NCHUNK 16
#define CHUNK_N (N_ / NCHUNK) // 576

// D(16x16 f32) = A(16x4 f32) * B(4x16 f32) + C
// A layout: lane m=lane&15 holds row m; VGPR0 = {K= (lane<16?0:2)}, VGPR1 = next K.
// B layout: lane holds col (lane&15); VGPR0 = {K = (lane<16?0:2)}, VGPR1 = next K.
// C/D layout: VGPR r, lane l -> M = r + 8*(l>=16), N = l&15.
static __device__ __forceinline__ v8f wmma4(v2f a, v2f b, v8f c) {
  return __builtin_amdgcn_wmma_f32_16x16x4_f32(false, a, false, b, (short)0, c,
                                               false, false);
}

// ---------------------------------------------------------------------------
// Projection: Y[b,o,n] = relu(scale[o] * (W[o,:] . X[b,:,n]) + shift[o])
// grid = (N/512, O/16, B), block = 256 (8 waves); each wave: 16(o) x 64(n)
// ---------------------------------------------------------------------------
__global__ __launch_bounds__(256) void proj_kernel(
    const float* __restrict__ X, const float* __restrict__ Wt,
    const float* __restrict__ gg, const float* __restrict__ bb,
    const float* __restrict__ mm, const float* __restrict__ vv,
    float* __restrict__ out, int O) {
  const int lane = threadIdx.x & 31;
  const int wave = threadIdx.x >> 5;
  const int m = lane & 15;
  const int hi = lane >> 4;
  const int koff = hi << 1;
  const int b = blockIdx.z;
  const int o0 = blockIdx.y * 16;
  const int n0 = (blockIdx.x * 8 + wave) * 64;

  const float* xb = X + (size_t)b * C_ * N_;
  const float* wrow = Wt + (size_t)(o0 + m) * C_;

  v8f acc0 = {}, acc1 = {}, acc2 = {}, acc3 = {};
  for (int c = 0; c < C_; c += 4) {
    v2f a = *(const v2f*)(wrow + c + koff);            // contiguous pair
    const float* xc0 = xb + (size_t)(c + koff) * N_ + n0 + m;
    const float* xc1 = xc0 + N_;
    v2f bv;
    bv.x = xc0[0];  bv.y = xc1[0];  acc0 = wmma4(a, bv, acc0);
    bv.x = xc0[16]; bv.y = xc1[16]; acc1 = wmma4(a, bv, acc1);
    bv.x = xc0[32]; bv.y = xc1[32]; acc2 = wmma4(a, bv, acc2);
    bv.x = xc0[48]; bv.y = xc1[48]; acc3 = wmma4(a, bv, acc3);
  }

  float* ob = out + (size_t)b * O * N_;
#pragma unroll
  for (int r = 0; r < 8; ++r) {
    int o = o0 + r + (hi << 3);
    float sc = gg[o] * rsqrtf(vv[o] + EPS_BN);
    float sh = bb[o] - mm[o] * sc;
    float* orow = ob + (size_t)o * N_ + n0 + m;
    orow[0]  = fmaxf(acc0[r] * sc + sh, 0.f);
    orow[16] = fmaxf(acc1[r] * sc + sh, 0.f);
    orow[32] = fmaxf(acc2[r] * sc + sh, 0.f);
    orow[48] = fmaxf(acc3[r] * sc + sh, 0.f);
  }
}

// ---------------------------------------------------------------------------
// Softmax along N (length 9216) for each (b,s) row; block per row.
// ---------------------------------------------------------------------------
__global__ __launch_bounds__(256) void softmax_rows_kernel(float* __restrict__ Y) {
  __shared__ float red[256];
  const int t = threadIdx.x;
  float* p = Y + (size_t)blockIdx.x * N_;
  float vals[36];
  float mx = -3.4e38f;
#pragma unroll
  for (int i = 0; i < 36; ++i) { vals[i] = p[t + (i << 8)]; mx = fmaxf(mx, vals[i]); }
  red[t] = mx; __syncthreads();
  for (int s = 128; s > 0; s >>= 1) { if (t < s) red[t] = fmaxf(red[t], red[t + s]); __syncthreads(); }
  mx = red[0]; __syncthreads();
  float sum = 0.f;
#pragma unroll
  for (int i = 0; i < 36; ++i) { vals[i] = expf(vals[i] - mx); sum += vals[i]; }
  red[t] = sum; __syncthreads();
  for (int s = 128; s > 0; s >>= 1) { if (t < s) red[t] += red[t + s]; __syncthreads(); }
  float inv = 1.f / red[0];
#pragma unroll
  for (int i = 0; i < 36; ++i) p[t + (i << 8)] = vals[i] * inv;
}

// ---------------------------------------------------------------------------
// Softmax along S (channels, stride N) for each (b,n); one thread per column.
// ---------------------------------------------------------------------------
__global__ __launch_bounds__(256) void softmax_cols_kernel(float* __restrict__ Y) {
  size_t idx = (size_t)blockIdx.x * 256 + threadIdx.x; // b*N_ + n
  int b = (int)(idx / N_);
  int n = (int)(idx - (size_t)b * N_);
  float* p = Y + (size_t)b * S_ * N_ + n;
  float mx = -3.4e38f;
  for (int s = 0; s < S_; ++s) mx = fmaxf(mx, p[(size_t)s * N_]);
  float sum = 0.f;
  for (int s = 0; s < S_; ++s) sum += expf(p[(size_t)s * N_] - mx);
  float inv = 1.f / sum;
  for (int s = 0; s < S_; ++s) p[(size_t)s * N_] = expf(p[(size_t)s * N_] - mx) * inv;
}

// ---------------------------------------------------------------------------
// discrib[b,k,s] = sum_n x_phi[b,k,n]*x_theta[b,s,n]
// grid = (4, 4, B), block = 128 (4 waves); wave -> one 16x16 (k,s) tile.
// ---------------------------------------------------------------------------
__global__ __launch_bounds__(128) void discrib_kernel(
    const float* __restrict__ xphi, const float* __restrict__ xth,
    float* __restrict__ dsc) {
  const int lane = threadIdx.x & 31;
  const int wave = threadIdx.x >> 5;
  const int m = lane & 15;
  const int hi = lane >> 4;
  const int koff = hi << 1;
  const int b = blockIdx.z;
  const int k0 = blockIdx.y * 16;
  const int s0 = (blockIdx.x * 4 + wave) * 16;
  const float* arow = xphi + ((size_t)b * K_ + k0 + m) * N_;
  const float* brow = xth + ((size_t)b * S_ + s0 + m) * N_;
  v8f acc = {};
  for (int n = 0; n < N_; n += 4) {
    v2f a = *(const v2f*)(arow + n + koff);
    v2f bv = *(const v2f*)(brow + n + koff);
    acc = wmma4(a, bv, acc);
  }
  float* dst = dsc + (size_t)b * K_ * S_;
#pragma unroll
  for (int r = 0; r < 8; ++r) {
    int k = k0 + r + (hi << 3);
    dst[(size_t)k * S_ + s0 + m] = acc[r];
  }
}

// ---------------------------------------------------------------------------
// Qraw[b,n,k] = sum_s discrib[b,k,s]*x_rou[b,s,n]
// grid = (N/16, B), block = 128; wave w covers k-tile w; LDS transpose for
// coalesced (n,k)-major stores.
// ---------------------------------------------------------------------------
__global__ __launch_bounds__(128) void qgemm_kernel(
    const float* __restrict__ dsc, const float* __restrict__ xrou,
    float* __restrict__ Qraw) {
  __shared__ float tile[64 * 16]; // [k][n_local]
  const int lane = threadIdx.x & 31;
  const int wave = threadIdx.x >> 5;
  const int m = lane & 15;
  const int hi = lane >> 4;
  const int koff = hi << 1;
  const int b = blockIdx.y;
  const int n0 = blockIdx.x * 16;
  const int k0 = wave * 16;
  const float* arow = dsc + ((size_t)b * K_ + k0 + m) * S_;
  const float* bbase = xrou + (size_t)b * S_ * N_ + n0 + m;
  v8f acc = {};
  for (int s = 0; s < S_; s += 4) {
    v2f a = *(const v2f*)(arow + s + koff);
    v2f bv;
    bv.x = bbase[(size_t)(s + koff) * N_];
    bv.y = bbase[(size_t)(s + koff + 1) * N_];
    acc = wmma4(a, bv, acc);
  }
#pragma unroll
  for (int r = 0; r < 8; ++r) tile[(k0 + r + (hi << 3)) * 16 + m] = acc[r];
  __syncthreads();
  const int t = threadIdx.x;
  const int nl = t >> 3;
  const int kp = (t & 7) * 8;
  float* qrow = Qraw + ((size_t)b * N_ + n0 + nl) * K_ + kp;
  v4f oA, oB;
  oA.x = tile[(kp + 0) * 16 + nl]; oA.y = tile[(kp + 1) * 16 + nl];
  oA.z = tile[(kp + 2) * 16 + nl]; oA.w = tile[(kp + 3) * 16 + nl];
  oB.x = tile[(kp + 4) * 16 + nl]; oB.y = tile[(kp + 5) * 16 + nl];
  oB.z = tile[(kp + 6) * 16 + nl]; oB.w = tile[(kp + 7) * 16 + nl];
  *(v4f*)qrow = oA;
  *(v4f*)(qrow + 4) = oB;
}

// ---------------------------------------------------------------------------
// Per-(b,k) column reduction over n. mode 0: 1/max(||.||2, eps); mode 1: sum.
// ---------------------------------------------------------------------------
__global__ __launch_bounds__(256) void colreduce_kernel(
    const float* __restrict__ Q, float* __restrict__ out, int mode) {
  __shared__ float red[256];
  const int t = threadIdx.x;
  const int bk = blockIdx.x;
  const int b = bk >> 6;
  const int k = bk & 63;
  const float* p = Q + (size_t)b * N_ * K_ + k;
  float s = 0.f;
  for (int n = t; n < N_; n += 256) {
    float q = p[(size_t)n * K_];
    s += mode ? q : q * q;
  }
  red[t] = s; __syncthreads();
  for (int st = 128; st > 0; st >>= 1) { if (t < st) red[t] += red[t + st]; __syncthreads(); }
  if (t == 0) out[bk] = mode ? red[0] : 1.f / fmaxf(sqrtf(red[0]), EPS_NORM);
}

// ---------------------------------------------------------------------------
// Q[b,n,:] = softmax_k(Qraw[b,n,k] * colfac[b,k]); thread per (b,n).
// ---------------------------------------------------------------------------
__global__ __launch_bounds__(256) void qnorm_softmax_kernel(
    float* __restrict__ Q, const float* __restrict__ colfac) {
  __shared__ float fac[64];
  size_t idx = (size_t)blockIdx.x * 256 + threadIdx.x; // b*N_ + n (block spans one b)
  int b = (int)(idx / N_);
  if (threadIdx.x < 64) fac[threadIdx.x] = colfac[b * 64 + threadIdx.x];
  __syncthreads();
  float* p = Q + idx * K_;
  float q[64];
  float mx = -3.4e38f;
#pragma unroll
  for (int k = 0; k < 64; ++k) { q[k] = p[k] * fac[k]; mx = fmaxf(mx, q[k]); }
  float sum = 0.f;
#pragma unroll
  for (int k = 0; k < 64; ++k) { q[k] = expf(q[k] - mx); sum += q[k]; }
  float inv = 1.f / sum;
#pragma unroll
  for (int k = 0; k < 64; ++k) p[k] = q[k] * inv;
}

// ---------------------------------------------------------------------------
// Znum partials: part[b,c,s,k] = sum_{n in chunk c} x_val[b,s,n]*Q[b,n,k]
// grid = (4, NCHUNK, B), block = 128; wave: 16(s) x 64(k).
// ---------------------------------------------------------------------------
__global__ __launch_bounds__(128) void znum_kernel(
    const float* __restrict__ xval, const float* __restrict__ Q,
    float* __restrict__ part) {
  const int lane = threadIdx.x & 31;
  const int wave = threadIdx.x >> 5;
  const int m = lane & 15;
  const int hi = lane >> 4;
  const int koff = hi << 1;
  const int b = blockIdx.z;
  const int chunk = blockIdx.y;
  const int s0 = (blockIdx.x * 4 + wave) * 16;
  const float* arow = xval + ((size_t)b * S_ + s0 + m) * N_;
  const float* qb = Q + (size_t)b * N_ * K_;
  v8f acc0 = {}, acc1 = {}, acc2 = {}, acc3 = {};
  const int nbeg = chunk * CHUNK_N;
  for (int n = nbeg; n < nbeg + CHUNK_N; n += 4) {
    v2f a = *(const v2f*)(arow + n + koff);
    const float* q0 = qb + (size_t)(n + koff) * K_ + m;
    const float* q1 = q0 + K_;
    v2f bv;
    bv.x = q0[0];  bv.y = q1[0];  acc0 = wmma4(a, bv, acc0);
    bv.x = q0[16]; bv.y = q1[16]; acc1 = wmma4(a, bv, acc1);
    bv.x = q0[32]; bv.y = q1[32]; acc2 = wmma4(a, bv, acc2);
    bv.x = q0[48]; bv.y = q1[48]; acc3 = wmma4(a, bv, acc3);
  }
  float* dst = part + ((size_t)b * NCHUNK + chunk) * S_ * K_;
#pragma unroll
  for (int r = 0; r < 8; ++r) {
    int s = s0 + r + (hi << 3);
    float* row = dst + (size_t)s * K_ + m;
    row[0] = acc0[r]; row[16] = acc1[r]; row[32] = acc2[r]; row[48] = acc3[r];
  }
}

// ---------------------------------------------------------------------------
// Z[b,s,k] = l2norm_s( (sum_c part[b,c,s,k]) / qsum[b,k] ); block per (b,k).
// ---------------------------------------------------------------------------
__global__ __launch_bounds__(256) void zfinal_kernel(
    const float* __restrict__ part, const float* __restrict__ qs,
    float* __restrict__ Z) {
  __shared__ float red[256];
  const int bk = blockIdx.x;
  const int b = bk >> 6;
  const int k = bk & 63;
  const int s = threadIdx.x;
  const float* p = part + (size_t)b * NCHUNK * S_ * K_ + (size_t)s * K_ + k;
  float z = 0.f;
#pragma unroll
  for (int c = 0; c < NCHUNK; ++c) z += p[(size_t)c * S_ * K_];
  z /= qs[bk];
  red[s] = z * z; __syncthreads();
  for (int st = 128; st > 0; st >>= 1) { if (s < st) red[s] += red[s + st]; __syncthreads(); }
  float fac = 1.f / fmaxf(sqrtf(red[0]), EPS_NORM);
  Z[((size_t)b * S_ + s) * K_ + k] = z * fac;
}

// ---------------------------------------------------------------------------
extern "C" void kernel_launch(void* const* d_in, const int* in_sizes, int n_in,
                              void* d_out, int out_size, void* d_ws, size_t ws_size,
                              hipStream_t stream) {
  (void)in_sizes; (void)n_in; (void)out_size; (void)ws_size;
  const float* X = (const float*)d_in[0];
  const float* phi_w = (const float*)d_in[1];
  const float* phi_g = (const float*)d_in[2];
  const float* phi_b = (const float*)d_in[3];
  const float* phi_m = (const float*)d_in[4];
  const float* phi_v = (const float*)d_in[5];
  const float* th_w = (const float*)d_in[6];
  const float* th_g = (const float*)d_in[7];
  const float* th_b = (const float*)d_in[8];
  const float* th_m = (const float*)d_in[9];
  const float* th_v = (const float*)d_in[10];
  const float* ro_w = (const float*)d_in[11];
  const float* ro_g = (const float*)d_in[12];
  const float* ro_b = (const float*)d_in[13];
  const float* ro_m = (const float*)d_in[14];
  const float* ro_v = (const float*)d_in[15];
  const float* va_w = (const float*)d_in[16];
  const float* va_g = (const float*)d_in[17];
  const float* va_b = (const float*)d_in[18];
  const float* va_m = (const float*)d_in[19];
  const float* va_v = (const float*)d_in[20];

  // Workspace layout (floats). bufA: x_theta then x_val. bufB: x_rou then
  // Znum partials (2.1M < 18.9M). Total ~42.6M floats (~170 MB).
  float* ws = (float*)d_ws;
  float* bufA = ws;
  float* bufB = bufA + (size_t)B_ * S_ * N_;
  float* xphi = bufB + (size_t)B_ * S_ * N_;
  float* dsc = xphi + (size_t)B_ * K_ * N_;
  float* colfac = dsc + (size_t)B_ * K_ * S_;
  float* qsum = colfac + B_ * K_;

  float* Z = (float*)d_out;                 // (B,S,K)
  float* Q = Z + (size_t)B_ * S_ * K_;      // (B,N,K)

  dim3 b256(256), b128(128);

  proj_kernel<<<dim3(N_ / 512, K_ / 16, B_), b256, 0, stream>>>(
      X, phi_w, phi_g, phi_b, phi_m, phi_v, xphi, K_);
  proj_kernel<<<dim3(N_ / 512, S_ / 16, B_), b256, 0, stream>>>(
      X, th_w, th_g, th_b, th_m, th_v, bufA, S_);
  softmax_rows_kernel<<<B_ * S_, b256, 0, stream>>>(bufA);
  proj_kernel<<<dim3(N_ / 512, S_ / 16, B_), b256, 0, stream>>>(
      X, ro_w, ro_g, ro_b, ro_m, ro_v, bufB, S_);
  softmax_cols_kernel<<<(B_ * N_) / 256, b256, 0, stream>>>(bufB);
  discrib_kernel<<<dim3(4, 4, B_), b128, 0, stream>>>(xphi, bufA, dsc);
  proj_kernel<<<dim3(N_ / 512, S_ / 16, B_), b256, 0, stream>>>(
      X, va_w, va_g, va_b, va_m, va_v, bufA, S_); // x_val overwrites x_theta
  qgemm_kernel<<<dim3(N_ / 16, B_), b128, 0, stream>>>(dsc, bufB, Q);
  colreduce_kernel<<<B_ * K_, b256, 0, stream>>>(Q, colfac, 0);
  qnorm_softmax_kernel<<<(B_ * N_) / 256, b256, 0, stream>>>(Q, colfac);
  colreduce_kernel<<<B_ * K_, b256, 0, stream>>>(Q, qsum, 1);
  znum_kernel<<<dim3(4, NCHUNK, B_), b128, 0, stream>>>(bufA, Q, bufB);
  zfinal_kernel<<<B_ * K_, b256, 0, stream>>>(bufB, qsum, Z);
}